// LMCritierion_35794257444926
// MI455X (gfx1250) — compile-verified
//
#include <hip/hip_runtime.h>
#include <math.h>

// ---------------------------------------------------------------------------
// Label-smoothed KL loss over (N=4096, V=32000) f32 logits.
// Memory-bound: 512 MB streamed once -> ~22 us @ 23.3 TB/s.
// One block per row, 8 wave32 waves, b128 non-temporal loads.
// Plain row-sum (Sigma x) is computed on the matrix pipe with
// V_WMMA_F32_16X16X4_F32 (A = loaded data, B = ones, C accumulates),
// co-executing with the VALU online-softmax exp chain.
// ---------------------------------------------------------------------------

typedef __attribute__((ext_vector_type(2))) float v2f;
typedef __attribute__((ext_vector_type(4))) float f32x4;
typedef __attribute__((ext_vector_type(8))) float v8f;

#define VOCAB   32000
#define NROWS   4096
#define VEC4S   (VOCAB / 4)          // 8000 float4 per row
#define MAINV4  7936                 // 31 * 256 -> uniform trip count
#define TAILV4  (VEC4S - MAINV4)     // 64, handled by waves 0..1 (full EXEC)
#define LS_P1   0.9f
#define LS_P2   3.125e-06f

__global__ __launch_bounds__(256)
void ls_kl_row_kernel(const float* __restrict__ x,
                      const int*   __restrict__ labels,
                      float*       __restrict__ row_loss) {
    const int row  = blockIdx.x;
    const int tid  = threadIdx.x;
    const int lane = tid & 31;
    const int wave = tid >> 5;

    const f32x4* __restrict__ rp = (const f32x4*)(x + (size_t)row * VOCAB);

    float m = -INFINITY;   // running max (online softmax)
    float s = 0.0f;        // running sum exp(x - m)
    float t_extra = 0.0f;  // tail plain-sum (VALU)
    v8f   acc = {};        // WMMA accumulator: D = A*ones + C  (row sums)
    v2f   ones; ones[0] = 1.0f; ones[1] = 1.0f;

    // Main body: 31 iterations for every thread -> wave-uniform, EXEC all-1s
    // (required for WMMA). Stride-256 float4 -> fully coalesced b128.
    for (int k = 0; k < 31; ++k) {
        f32x4 v = __builtin_nontemporal_load(&rp[tid + (k << 8)]);

        // --- online softmax partial (VALU + trans units) ---
        float vm = fmaxf(fmaxf(v.x, v.y), fmaxf(v.z, v.w));
        float nm = fmaxf(m, vm);
        float r  = __expf(m - nm);     // rescale old sum (exp(-inf)=0 first iter)
        s = s * r + __expf(v.x - nm) + __expf(v.y - nm)
                  + __expf(v.z - nm) + __expf(v.w - nm);
        m = nm;

        // --- plain Sigma x on the matrix pipe ---
        // Any bijection of the wave's 128 floats onto two 16x4 A tiles works,
        // since D[m][n] = Sigma_k A[m][k] * 1 and we only need the total.
        v2f a0; a0[0] = v.x; a0[1] = v.y;
        v2f a1; a1[0] = v.z; a1[1] = v.w;
        acc = __builtin_amdgcn_wmma_f32_16x16x4_f32(false, a0, false, ones,
                                                    (short)0, acc, false, false);
        acc = __builtin_amdgcn_wmma_f32_16x16x4_f32(false, a1, false, ones,
                                                    (short)0, acc, false, false);
    }

    // Tail: 64 remaining float4 -> waves 0 and 1 execute fully (uniform per
    // wave). VALU-only here (no WMMA under a potentially masked branch).
    if (tid < TAILV4) {
        f32x4 v = __builtin_nontemporal_load(&rp[MAINV4 + tid]);
        float vm = fmaxf(fmaxf(v.x, v.y), fmaxf(v.z, v.w));
        float nm = fmaxf(m, vm);
        float r  = __expf(m - nm);
        s = s * r + __expf(v.x - nm) + __expf(v.y - nm)
                  + __expf(v.z - nm) + __expf(v.w - nm);
        m = nm;
        t_extra += (v.x + v.y) + (v.z + v.w);
    }

    // ---- wave-level reduction (wave32 xor-shuffles) ----
    for (int off = 16; off > 0; off >>= 1) {
        float mo = __shfl_xor(m, off, 32);
        float so = __shfl_xor(s, off, 32);
        float to = __shfl_xor(t_extra, off, 32);
        float nm = fmaxf(m, mo);
        s = s * __expf(m - nm) + so * __expf(mo - nm);
        m = nm;
        t_extra += to;
    }
    // WMMA accumulator layout: VGPR r holds D[M=r][N=lane] (lanes 0-15) and
    // D[M=r+8][N=lane-16] (lanes 16-31). All N columns equal (B is ones), so
    // per-lane sum of the 8 VGPRs + a lane 0<->16 fold yields the wave total.
    float tl = ((acc[0] + acc[1]) + (acc[2] + acc[3]))
             + ((acc[4] + acc[5]) + (acc[6] + acc[7]));
    tl += __shfl_xor(tl, 16, 32);

    __shared__ float m_arr[8], s_arr[8], t_arr[8];
    if (lane == 0) {
        m_arr[wave] = m;
        s_arr[wave] = s;
        t_arr[wave] = tl + t_extra;
    }
    __syncthreads();

    if (tid == 0) {
        float M = m_arr[0], S = s_arr[0], T = t_arr[0];
        for (int w = 1; w < 8; ++w) {
            float nm = fmaxf(M, m_arr[w]);
            S = S * __expf(M - nm) + s_arr[w] * __expf(m_arr[w] - nm);
            M = nm;
            T += t_arr[w];
        }
        float lse   = M + __logf(S);
        int   lab   = labels[row];
        float xl    = x[(size_t)row * VOCAB + lab];
        float s_lab = xl - lse;                       // scores[label]
        float rs    = T - (float)VOCAB * lse;         // Sigma scores
        const float cterm = LS_P1 * __logf(LS_P1)
                          + (float)(VOCAB - 1) * LS_P2 * __logf(LS_P2);
        row_loss[row] = cterm - LS_P1 * s_lab - LS_P2 * (rs - s_lab);
    }
}

__global__ __launch_bounds__(256)
void ls_kl_reduce_kernel(const float* __restrict__ row_loss,
                         float*       __restrict__ out) {
    __shared__ float sm[256];
    float t = 0.0f;
    for (int i = threadIdx.x; i < NROWS; i += 256) t += row_loss[i];
    sm[threadIdx.x] = t;
    __syncthreads();
    for (int w = 128; w > 0; w >>= 1) {
        if (threadIdx.x < w) sm[threadIdx.x] += sm[threadIdx.x + w];
        __syncthreads();
    }
    if (threadIdx.x == 0) out[0] = sm[0];
}

extern "C" void kernel_launch(void* const* d_in, const int* in_sizes, int n_in,
                              void* d_out, int out_size, void* d_ws, size_t ws_size,
                              hipStream_t stream) {
    const float* dec_outs = (const float*)d_in[0];
    const int*   labels   = (const int*)d_in[1];
    float*       row_loss = (float*)d_ws;     // 4096 floats of scratch

    ls_kl_row_kernel<<<NROWS, 256, 0, stream>>>(dec_outs, labels, row_loss);
    ls_kl_reduce_kernel<<<1, 256, 0, stream>>>(row_loss, (float*)d_out);
}